// Cond_PlanarTrans_84765474554410
// MI455X (gfx1250) — compile-verified
//
#include <hip/hip_runtime.h>

// Problem sizes (fixed by the reference).
#define B_SZ 128
#define P_SZ 2048
#define DIM  64
#define N_M  8
#define HID  (N_M * DIM)   // 512

typedef float v2f __attribute__((ext_vector_type(2)));
typedef float v4f __attribute__((ext_vector_type(4)));
typedef float v8f __attribute__((ext_vector_type(8)));

// ---------------------------------------------------------------------------
// Kernel 1: Y = relu(o @ W^T + b)  for W1->w and W2->u, via V_WMMA_F32_16X16X4_F32.
// One wave (32 lanes) per 16x16 output tile. grid = (512/16, 128/16, 2).
//
// A tile (16x4 f32, ISA 7.12.2): lane<16 holds A[lane][k0..k0+1] in {v0,v1};
// lane>=16 holds A[lane-16][k0+2..k0+3]. B tile (4x16) mirrors with N across
// lanes: lane holds B[k0+2*half .. +1][n0+(lane&15)] = W[n0+(lane&15)][k...]
// since B = W^T. Both loads are therefore identical row-major/K-fast reads.
// C/D: vgpr v, lane L -> row m0+v+8*(L>>4), col n0+(L&15).
// ---------------------------------------------------------------------------
__global__ __launch_bounds__(32) void cond_mlp_wmma(
    const float* __restrict__ o,    // [128,64]
    const float* __restrict__ W1, const float* __restrict__ b1,  // [512,64],[512]
    const float* __restrict__ W2, const float* __restrict__ b2,
    float* __restrict__ outW,       // [128,512] relu'd
    float* __restrict__ outU)       // [128,512] relu'd
{
    const int lane = threadIdx.x;          // 0..31, all active (EXEC all-1s)
    const int n0   = blockIdx.x * 16;      // output-feature tile
    const int m0   = blockIdx.y * 16;      // batch-row tile
    const float* W    = (blockIdx.z == 0) ? W1 : W2;
    const float* bias = (blockIdx.z == 0) ? b1 : b2;
    float*       out  = (blockIdx.z == 0) ? outW : outU;

    const int l16  = lane & 15;
    const int half = lane >> 4;

    const float* arow = o + (size_t)(m0 + l16) * DIM;   // A: rows of o
    const float* brow = W + (size_t)(n0 + l16) * DIM;   // B: rows of W (= cols of W^T)

    v8f c = {};
#pragma unroll
    for (int k0 = 0; k0 < DIM; k0 += 4) {
        const int kb = k0 + half * 2;
        v2f a, b;
        a.x = arow[kb]; a.y = arow[kb + 1];
        b.x = brow[kb]; b.y = brow[kb + 1];
        // (neg_a, A, neg_b, B, c_mod, C, reuse_a, reuse_b)
        c = __builtin_amdgcn_wmma_f32_16x16x4_f32(false, a, false, b,
                                                  (short)0, c, false, false);
    }

    const int   n  = n0 + l16;
    const float bn = bias[n];
#pragma unroll
    for (int v = 0; v < 8; ++v) {
        const int mrow = m0 + v + half * 8;
        float y = c[v] + bn;
        out[(size_t)mrow * HID + n] = (y > 0.0f) ? y : 0.0f;
    }
}

// ---------------------------------------------------------------------------
// Kernel 1b: bflow = relu(o @ W3^T + b3) : [128, 8]. Trivial (1024 dots of 64).
// ---------------------------------------------------------------------------
__global__ __launch_bounds__(256) void cond_bflow(
    const float* __restrict__ o,   // [128,64]
    const float* __restrict__ W3,  // [8,64]
    const float* __restrict__ b3,  // [8]
    float* __restrict__ outBF)     // [128,8]
{
    const int t = blockIdx.x * 256 + threadIdx.x;
    if (t >= B_SZ * N_M) return;
    const int b = t >> 3, j = t & 7;
    const float* orow = o + (size_t)b * DIM;
    const float* wrow = W3 + (size_t)j * DIM;
    float s = b3[j];
#pragma unroll
    for (int k = 0; k < DIM; ++k) s = fmaf(orow[k], wrow[k], s);
    outBF[t] = (s > 0.0f) ? s : 0.0f;
}

// ---------------------------------------------------------------------------
// Kernel 2: streaming planar flow (HBM-bound: 64MB in + 64MB out).
// grid = (P/128, B), block = 256 (8 wave32).
// 16 lanes own one particle (float4/lane = 64 dims), 2 particles per wave.
// Per-batch tables w[8][64], u[8][64], b[8] staged in LDS (one fill / block).
// Non-temporal 128b loads/stores: data is touched exactly once.
// ---------------------------------------------------------------------------
__global__ __launch_bounds__(256) void planar_flow_apply(
    const int*   __restrict__ m,     // [128,2048]
    const float* __restrict__ s_t,   // [128,2048,64]
    const float* __restrict__ wbuf,  // [128,512]
    const float* __restrict__ ubuf,  // [128,512]
    const float* __restrict__ bfbuf, // [128,8]
    float* __restrict__ out)         // [128,2048,64]
{
    __shared__ float lw[HID];
    __shared__ float lu[HID];
    __shared__ float lbf[N_M];

    const int b   = blockIdx.y;
    const int tid = threadIdx.x;

    for (int i = tid; i < HID; i += 256) {
        lw[i] = wbuf[(size_t)b * HID + i];
        lu[i] = ubuf[(size_t)b * HID + i];
    }
    if (tid < N_M) lbf[tid] = bfbuf[b * N_M + tid];
    __syncthreads();

    const int lane = tid & 31;
    const int wv   = tid >> 5;        // wave in block: 0..7
    const int half = lane >> 4;       // particle within wave: 0/1
    const int l16  = lane & 15;       // dim-quad within particle

    const int    pbase = blockIdx.x * 128;
    const size_t srow  = (size_t)b * P_SZ;

#pragma unroll
    for (int it = 0; it < 8; ++it) {
        const int    p   = pbase + it * 16 + wv * 2 + half;
        const size_t off = (srow + p) * DIM + l16 * 4;

        const v4f s4 = __builtin_nontemporal_load((const v4f*)(s_t + off));
        const int mm = m[srow + p];                 // same addr across 16 lanes

        const v4f w4 = *(const v4f*)(lw + mm * DIM + l16 * 4);
        const v4f u4 = *(const v4f*)(lu + mm * DIM + l16 * 4);

        float part = s4.x * w4.x + s4.y * w4.y + s4.z * w4.z + s4.w * w4.w;
        // reduce across the 16-lane group (xor masks < 16 stay in-half, wave32)
        part += __shfl_xor(part, 1);
        part += __shfl_xor(part, 2);
        part += __shfl_xor(part, 4);
        part += __shfl_xor(part, 8);

        const float t  = tanhf(part + lbf[mm]);
        const v4f  r4  = s4 + u4 * t;
        __builtin_nontemporal_store(r4, (v4f*)(out + off));
    }
}

// ---------------------------------------------------------------------------
extern "C" void kernel_launch(void* const* d_in, const int* in_sizes, int n_in,
                              void* d_out, int out_size, void* d_ws, size_t ws_size,
                              hipStream_t stream) {
    const int*   m   = (const int*)  d_in[0];
    const float* s_t = (const float*)d_in[1];
    const float* o   = (const float*)d_in[2];
    const float* W1  = (const float*)d_in[3];
    const float* b1  = (const float*)d_in[4];
    const float* W2  = (const float*)d_in[5];
    const float* b2  = (const float*)d_in[6];
    const float* W3  = (const float*)d_in[7];
    const float* b3  = (const float*)d_in[8];
    float*       out = (float*)d_out;

    float* wbuf  = (float*)d_ws;                 // [128,512]
    float* ubuf  = wbuf + (size_t)B_SZ * HID;    // [128,512]
    float* bfbuf = ubuf + (size_t)B_SZ * HID;    // [128,8]   (total ~528 KB)

    // Stage 1: conditioning MLP via f32 WMMA (2 matrices x 8x32 tiles).
    cond_mlp_wmma<<<dim3(HID / 16, B_SZ / 16, 2), 32, 0, stream>>>(
        o, W1, b1, W2, b2, wbuf, ubuf);
    cond_bflow<<<(B_SZ * N_M + 255) / 256, 256, 0, stream>>>(o, W3, b3, bfbuf);

    // Stage 2: HBM-bound streaming apply.
    planar_flow_apply<<<dim3(P_SZ / 128, B_SZ), 256, 0, stream>>>(
        m, s_t, wbuf, ubuf, bfbuf, out);
}